// SSIM_43568148251078
// MI455X (gfx1250) — compile-verified
//
#include <hip/hip_runtime.h>

// SSIM (11x11 box window, count_include_pad) over 16x3x512x512 f32 tensors.
// Memory-bound: ~100.7 MB read => ~4.3 us floor at 23.3 TB/s.
// Vertical box sum done as banded-matrix WMMA: V_WMMA_F32_16X16X4_F32.

typedef float v2f __attribute__((ext_vector_type(2)));
typedef float v8f __attribute__((ext_vector_type(8)));

#define WIN   11
#define PADR  5
#define TILE  32          // output tile per block (32x32)
#define REG   42          // TILE + WIN - 1 (halo region)
#define XSTR  44          // padded LDS row stride for raw inputs
#define IMGW  512
#define NIMG  48          // 16 * 3 planes
#define NBLK  (NIMG * 16 * 16)   // 12288 partials

__global__ __launch_bounds__(128)
void ssim_tile_kernel(const float* __restrict__ xg,
                      const float* __restrict__ yg,
                      float* __restrict__ partials)
{
    __shared__ float Xs[REG * XSTR];
    __shared__ float Ys[REG * XSTR];
    __shared__ float Hs[5][REG * TILE];   // horizontal box sums of 5 products
    __shared__ float red[128];

    const int tid = threadIdx.x;
    const int R0  = blockIdx.y * TILE;
    const int C0  = blockIdx.x * TILE;
    const long plane = (long)blockIdx.z * (IMGW * IMGW);
    const float* __restrict__ xp = xg + plane;
    const float* __restrict__ yp = yg + plane;

    // ---- Stage 1: zero-padded 42x42 halo load of both tensors ----
    for (int idx = tid; idx < REG * REG; idx += 128) {
        int rr = idx / REG, cc = idx - rr * REG;
        int gr = R0 - PADR + rr, gc = C0 - PADR + cc;
        bool ok = (gr >= 0) && (gr < IMGW) && (gc >= 0) && (gc < IMGW);
        int off = gr * IMGW + gc;
        Xs[rr * XSTR + cc] = ok ? xp[off] : 0.0f;
        Ys[rr * XSTR + cc] = ok ? yp[off] : 0.0f;
    }
    __syncthreads();

    // ---- Stage 2: horizontal 11-tap sums of x, y, x^2, y^2, x*y ----
    for (int idx = tid; idx < REG * TILE; idx += 128) {
        int rr = idx / TILE, c = idx - rr * TILE;
        float hx = 0.f, hy = 0.f, hxx = 0.f, hyy = 0.f, hxy = 0.f;
#pragma unroll
        for (int j = 0; j < WIN; ++j) {
            float x = Xs[rr * XSTR + c + j];
            float y = Ys[rr * XSTR + c + j];
            hx += x; hy += y;
            hxx = fmaf(x, x, hxx);
            hyy = fmaf(y, y, hyy);
            hxy = fmaf(x, y, hxy);
        }
        Hs[0][idx] = hx;  Hs[1][idx] = hy;
        Hs[2][idx] = hxx; Hs[3][idx] = hyy; Hs[4][idx] = hxy;
    }
    __syncthreads();

    // ---- Stage 3: vertical 11-tap sums via f32 WMMA banded matmul ----
    // Wave w handles the 16x16 sub-tile (ty = w>>1, tx = w&1).
    // D[i][n] = sum_{k'=i..i+10} Hs[ch][16*ty + k'][16*tx + n]
    //         = Band(16x26) x H(26x16), K padded to 28 -> 7 WMMA steps.
    const int lane = tid & 31;
    const int wave = tid >> 5;
    const int ty   = wave >> 1;
    const int tx   = wave & 1;
    const int half = lane >> 4;     // ISA: lanes 16-31 carry the upper K pair / M+8
    const int nl   = lane & 15;     // A: row M;  B/C/D: column N
    const int hbase = 16 * ty;      // Hs row of k' == 0 for this wave
    const int cbase = 16 * tx + nl; // Hs column for B operand lanes

    v8f acc[5];
#pragma unroll
    for (int ch = 0; ch < 5; ++ch) acc[ch] = (v8f){0,0,0,0,0,0,0,0};

#pragma unroll
    for (int kk = 0; kk < 7; ++kk) {
        // A (16x4 f32 layout): VGPR j holds K = 4*kk + j + 2*half, M = nl.
        v2f a;
#pragma unroll
        for (int j = 0; j < 2; ++j) {
            int kp = 4 * kk + j + 2 * half;
            a[j] = (kp >= nl && kp <= nl + 10) ? 1.0f : 0.0f;  // band of ones
        }
#pragma unroll
        for (int ch = 0; ch < 5; ++ch) {
            // B (4x16 f32 layout): VGPR j holds row K = 4*kk + j + 2*half, col N = nl.
            v2f b;
#pragma unroll
            for (int j = 0; j < 2; ++j) {
                int kp = 4 * kk + j + 2 * half;
                int rr = hbase + kp;
                if (rr > REG - 1) rr = REG - 1;          // clamp (value masked below)
                float v = Hs[ch][rr * TILE + cbase];
                b[j] = (kp < 26) ? v : 0.0f;             // keep NaN-free pad lanes
            }
            acc[ch] = __builtin_amdgcn_wmma_f32_16x16x4_f32(
                false, a, false, b, (short)0, acc[ch], false, false);
        }
    }

    // ---- Stage 4: elementwise SSIM on matching C/D layouts ----
    const float invw = 1.0f / (float)(WIN * WIN);
    const float C1 = 1e-4f;     // (0.01*1.0)^2
    const float C2 = 9e-4f;     // (0.03*1.0)^2
    const float EPSf = 1e-8f;
    float lsum = 0.0f;
#pragma unroll
    for (int e = 0; e < 8; ++e) {
        float mu1 = acc[0][e] * invw;
        float mu2 = acc[1][e] * invw;
        float s1  = acc[2][e] * invw - mu1 * mu1;
        float s2  = acc[3][e] * invw - mu2 * mu2;
        float s12 = acc[4][e] * invw - mu1 * mu2;
        float num = (2.0f * mu1 * mu2 + C1) * (2.0f * s12 + C2);
        float den = (mu1 * mu1 + mu2 * mu2 + C1) * (s1 + s2 + C2);
        lsum += num / (den + EPSf);
    }

    // ---- fixed-order block reduction -> one partial per block (deterministic) ----
    red[tid] = lsum;
    __syncthreads();
    for (int s = 64; s > 0; s >>= 1) {
        if (tid < s) red[tid] += red[tid + s];
        __syncthreads();
    }
    if (tid == 0) {
        int bflat = (blockIdx.z * gridDim.y + blockIdx.y) * gridDim.x + blockIdx.x;
        partials[bflat] = red[0];
    }
}

__global__ __launch_bounds__(256)
void ssim_reduce_kernel(const float* __restrict__ partials, int n,
                        float* __restrict__ out)
{
    __shared__ double red[256];
    double local = 0.0;
    for (int i = threadIdx.x; i < n; i += 256) local += (double)partials[i];
    red[threadIdx.x] = local;
    __syncthreads();
    for (int s = 128; s > 0; s >>= 1) {
        if (threadIdx.x < s) red[threadIdx.x] += red[threadIdx.x + s];
        __syncthreads();
    }
    if (threadIdx.x == 0) {
        double mean = red[0] / (double)(16.0 * 3.0 * 512.0 * 512.0);
        out[0] = (float)(1.0 - mean);
    }
}

extern "C" void kernel_launch(void* const* d_in, const int* in_sizes, int n_in,
                              void* d_out, int out_size, void* d_ws, size_t ws_size,
                              hipStream_t stream) {
    const float* y_pred = (const float*)d_in[0];
    const float* y_true = (const float*)d_in[1];
    float* partials = (float*)d_ws;          // NBLK floats, fully overwritten each call
    float* out = (float*)d_out;

    dim3 grid(16, 16, NIMG);                 // 512/32 x 512/32 tiles, 48 planes
    dim3 block(128);                         // 4 waves of 32
    ssim_tile_kernel<<<grid, block, 0, stream>>>(y_pred, y_true, partials);
    ssim_reduce_kernel<<<1, 256, 0, stream>>>(partials, NBLK, out);
}